// S_ATT_2388001816918
// MI455X (gfx1250) — compile-verified
//
#include <hip/hip_runtime.h>
#include <hip/hip_bf16.h>

typedef float v2f __attribute__((ext_vector_type(2)));
typedef float v8f __attribute__((ext_vector_type(8)));

namespace {
constexpr int Bn = 512;          // batch
constexpr int Sn = 256;          // sequence
constexpr int Dn = 256;          // model dim (= 2*emb)
constexpr int ROWS = 16;         // query-row tile per workgroup
constexpr int LDP = Dn + 4;      // LDS pitch (floats); 260 -> 16 rows on distinct banks
constexpr int NTHREADS = 256;    // 8 wave32
constexpr float LN_EPS = 1e-12f;
}

__device__ __forceinline__ v8f wmma4(v2f a, v2f b, v8f c) {
  // D(16x16,f32) = A(16x4,f32) * B(4x16,f32) + C
  return __builtin_amdgcn_wmma_f32_16x16x4_f32(false, a, false, b, (short)0, c,
                                               false, false);
}

__device__ __forceinline__ float wave_sum(float v) {
#pragma unroll
  for (int m = 16; m >= 1; m >>= 1) v += __shfl_xor(v, m, 32);
  return v;
}

__device__ __forceinline__ float wave_max(float v) {
#pragma unroll
  for (int m = 16; m >= 1; m >>= 1) v = fmaxf(v, __shfl_xor(v, m, 32));
  return v;
}

template <bool SQ>
__device__ __forceinline__ float powp(float x, float inv) {
  // x >= 0 ; x^inv. SQ fast path for alpha==1.5 (inv==2).
  return SQ ? x * x : exp2f(inv * log2f(x));  // log2(0)=-inf -> exp2 -> 0
}

// alpha-entmax via bisection over two rows (row = 2*wave+rr) of a 16 x Sn tile.
template <bool SQ>
__device__ __forceinline__ void entmax2(float* sS, int wave, int lane,
                                        float am1, float inv) {
  float Xs[2][8];
  float tau_lo[2], tau_m[2], dm[2], flo[2];
#pragma unroll
  for (int rr = 0; rr < 2; ++rr) {
    const int r = 2 * wave + rr;
    float mx = -3.4e38f;
#pragma unroll
    for (int e = 0; e < 8; ++e) {
      const float x = sS[r * LDP + e * 32 + lane] * am1;
      Xs[rr][e] = x;
      mx = fmaxf(mx, x);
    }
    mx = wave_max(mx);
    tau_lo[rr] = mx - 1.0f;
    const float tau_hi = mx - exp2f(am1 * log2f(1.0f / (float)Sn));
    float s = 0.f;
#pragma unroll
    for (int e = 0; e < 8; ++e)
      s += powp<SQ>(fmaxf(Xs[rr][e] - tau_lo[rr], 0.f), inv);
    flo[rr] = wave_sum(s) - 1.0f;
    dm[rr] = tau_hi - tau_lo[rr];
    tau_m[rr] = tau_lo[rr];
  }
  for (int it = 0; it < 50; ++it) {
    float fm[2];
#pragma unroll
    for (int rr = 0; rr < 2; ++rr) {          // two rows in flight for ILP
      dm[rr] *= 0.5f;
      tau_m[rr] = tau_lo[rr] + dm[rr];
      float s = 0.f;
#pragma unroll
      for (int e = 0; e < 8; ++e)
        s += powp<SQ>(fmaxf(Xs[rr][e] - tau_m[rr], 0.f), inv);
      fm[rr] = s;
    }
#pragma unroll
    for (int rr = 0; rr < 2; ++rr) {
      fm[rr] = wave_sum(fm[rr]) - 1.0f;
      if (fm[rr] * flo[rr] >= 0.f) tau_lo[rr] = tau_m[rr];
    }
  }
#pragma unroll
  for (int rr = 0; rr < 2; ++rr) {
    const int r = 2 * wave + rr;
    float p[8], s = 0.f;
#pragma unroll
    for (int e = 0; e < 8; ++e) {
      p[e] = powp<SQ>(fmaxf(Xs[rr][e] - tau_m[rr], 0.f), inv);
      s += p[e];
    }
    const float rs = 1.0f / wave_sum(s);
#pragma unroll
    for (int e = 0; e < 8; ++e) sS[r * LDP + e * 32 + lane] = p[e] * rs;
  }
}

// 16x32 output slice: A (16 x 256) from LDS, B fragments supplied by loadB(t, kl)
template <typename FB>
__device__ __forceinline__ void gemm2(const float* As, int lo, int hi, FB loadB,
                                      v8f& acc0, v8f& acc1) {
#pragma unroll 4
  for (int k0 = 0; k0 < Dn; k0 += 4) {   // K == 256 for every GEMM here
    const int kl = k0 + 2 * hi;
    const v2f a = *(const v2f*)(As + lo * LDP + kl);   // ds_load_b64
    const v2f vb0 = loadB(0, kl);
    const v2f vb1 = loadB(16, kl);
    acc0 = wmma4(a, vb0, acc0);
    acc1 = wmma4(a, vb1, acc1);
  }
}

__global__ __launch_bounds__(NTHREADS)
void fused_entmax_attention(const float* __restrict__ X,
                            const float* __restrict__ alpha_p,
                            const float* __restrict__ Wq,
                            const float* __restrict__ bq,
                            const float* __restrict__ W1,
                            const float* __restrict__ b1,
                            const float* __restrict__ W2,
                            const float* __restrict__ b2,
                            const float* __restrict__ gamma,
                            const float* __restrict__ beta,
                            float* __restrict__ out) {
  // three aliased stage buffers (48.75 KB total):
  __shared__ float bufA[ROWS * LDP];  // X rows            -> res
  __shared__ float bufB[ROWS * LDP];  // Q                 -> t = relu(res W1^T+b1)
  __shared__ float bufC[ROWS * LDP];  // scores -> attn    -> h + res

  const int b = blockIdx.y;
  const int i0 = blockIdx.x * ROWS;
  const int tid = threadIdx.x;
  const int wave = tid >> 5;
  const int lane = tid & 31;
  const int lo = lane & 15;   // frag row/col within 16
  const int hi = lane >> 4;   // frag half select
  const int cb = wave * 32;   // this wave's 32-column slice

  const float* __restrict__ Xb = X + (size_t)b * Sn * Dn;

  // ---- stage the 16 X rows of this tile into LDS ----
  for (int idx = tid; idx < ROWS * (Dn / 4); idx += NTHREADS) {
    const int r = idx / (Dn / 4);
    const int c = (idx - r * (Dn / 4)) * 4;
    *(float4*)&bufA[r * LDP + c] = *(const float4*)(Xb + (size_t)(i0 + r) * Dn + c);
  }
  __syncthreads();

  // ---- Q = relu(X Wq^T + bq) : bufA -> bufB ----
  {
    v8f c0 = {}, c1 = {};
    gemm2(bufA, lo, hi,
          [&](int t, int kl) {
            return *(const v2f*)(Wq + (size_t)(cb + t + lo) * Dn + kl);
          },
          c0, c1);
    const float q0 = bq[cb + lo], q1 = bq[cb + 16 + lo];
#pragma unroll
    for (int v = 0; v < 8; ++v) {
      const int r = v + 8 * hi;
      bufB[r * LDP + cb + lo] = fmaxf(c0[v] + q0, 0.f);
      bufB[r * LDP + cb + 16 + lo] = fmaxf(c1[v] + q1, 0.f);
    }
  }
  __syncthreads();

  // ---- scores = (Q X^T) / sqrt(D) : bufB -> bufC ----
  {
    v8f c0 = {}, c1 = {};
    gemm2(bufB, lo, hi,
          [&](int t, int kl) {
            return *(const v2f*)(Xb + (size_t)(cb + t + lo) * Dn + kl);
          },
          c0, c1);
    const float scl = rsqrtf((float)Dn);
#pragma unroll
    for (int v = 0; v < 8; ++v) {
      const int r = v + 8 * hi;
      bufC[r * LDP + cb + lo] = c0[v] * scl;
      bufC[r * LDP + cb + 16 + lo] = c1[v] * scl;
    }
  }
  __syncthreads();

  // ---- alpha-entmax (50-iter bisection) in place on bufC ----
  {
    const float am1 = alpha_p[0] - 1.0f;
    const float inv = 1.0f / am1;
    if (inv == 2.0f) entmax2<true>(bufC, wave, lane, am1, inv);
    else             entmax2<false>(bufC, wave, lane, am1, inv);
  }
  __syncthreads();

  // ---- res = attn @ X : bufC -> bufA ----
  {
    v8f c0 = {}, c1 = {};
#pragma unroll 2
    for (int k0 = 0; k0 < Sn; k0 += 4) {
      const int kl = k0 + 2 * hi;
      const v2f a = *(const v2f*)(bufC + lo * LDP + kl);
      v2f vb0, vb1;               // B[k][d] = X[b][k][d] (stride-D in k)
      vb0.x = Xb[(size_t)kl * Dn + cb + lo];
      vb0.y = Xb[(size_t)(kl + 1) * Dn + cb + lo];
      vb1.x = Xb[(size_t)kl * Dn + cb + 16 + lo];
      vb1.y = Xb[(size_t)(kl + 1) * Dn + cb + 16 + lo];
      c0 = wmma4(a, vb0, c0);
      c1 = wmma4(a, vb1, c1);
    }
#pragma unroll
    for (int v = 0; v < 8; ++v) {
      const int r = v + 8 * hi;
      bufA[r * LDP + cb + lo] = c0[v];
      bufA[r * LDP + cb + 16 + lo] = c1[v];
    }
  }
  __syncthreads();

  // ---- t = relu(res W1^T + b1) : bufA -> bufB ----
  {
    v8f c0 = {}, c1 = {};
    gemm2(bufA, lo, hi,
          [&](int t, int kl) {
            return *(const v2f*)(W1 + (size_t)(cb + t + lo) * Dn + kl);
          },
          c0, c1);
    const float q0 = b1[cb + lo], q1 = b1[cb + 16 + lo];
#pragma unroll
    for (int v = 0; v < 8; ++v) {
      const int r = v + 8 * hi;
      bufB[r * LDP + cb + lo] = fmaxf(c0[v] + q0, 0.f);
      bufB[r * LDP + cb + 16 + lo] = fmaxf(c1[v] + q1, 0.f);
    }
  }
  __syncthreads();

  // ---- h + res = t W2^T + b2 + res : bufB (+bufA) -> bufC ----
  {
    v8f c0 = {}, c1 = {};
    gemm2(bufB, lo, hi,
          [&](int t, int kl) {
            return *(const v2f*)(W2 + (size_t)(cb + t + lo) * Dn + kl);
          },
          c0, c1);
    const float q0 = b2[cb + lo], q1 = b2[cb + 16 + lo];
#pragma unroll
    for (int v = 0; v < 8; ++v) {
      const int r = v + 8 * hi;
      bufC[r * LDP + cb + lo]      = c0[v] + q0 + bufA[r * LDP + cb + lo];
      bufC[r * LDP + cb + 16 + lo] = c1[v] + q1 + bufA[r * LDP + cb + 16 + lo];
    }
  }
  __syncthreads();

  // ---- LayerNorm rows + scatter to split output layout ----
#pragma unroll
  for (int rr = 0; rr < 2; ++rr) {
    const int r = 2 * wave + rr;
    float x[8];
    float s = 0.f;
#pragma unroll
    for (int e = 0; e < 8; ++e) {
      x[e] = bufC[r * LDP + e * 32 + lane];
      s += x[e];
    }
    const float mu = wave_sum(s) * (1.0f / (float)Dn);
    float vv = 0.f;
#pragma unroll
    for (int e = 0; e < 8; ++e) {
      const float d = x[e] - mu;
      vv += d * d;
    }
    const float rstd = rsqrtf(wave_sum(vv) * (1.0f / (float)Dn) + LN_EPS);
    const int srow = i0 + r;
    float* dst = (srow < Sn - 1)
        ? out + (size_t)b * (Sn - 1) * Dn + (size_t)srow * Dn       // C_hat[:, :-1, :]
        : out + (size_t)Bn * (Sn - 1) * Dn + (size_t)b * Dn;        // C_hat[:, -1, :]
#pragma unroll
    for (int e = 0; e < 8; ++e) {
      const int col = e * 32 + lane;
      dst[col] = (x[e] - mu) * rstd * gamma[col] + beta[col];
    }
  }
}

extern "C" void kernel_launch(void* const* d_in, const int* in_sizes, int n_in,
                              void* d_out, int out_size, void* d_ws, size_t ws_size,
                              hipStream_t stream) {
  (void)in_sizes; (void)n_in; (void)out_size; (void)d_ws; (void)ws_size;
  dim3 grid(Sn / ROWS, Bn);   // (16, 512) tiles, 256 threads = 8 wave32 each
  hipLaunchKernelGGL(fused_entmax_attention, grid, dim3(NTHREADS), 0, stream,
                     (const float*)d_in[0], (const float*)d_in[1],
                     (const float*)d_in[2], (const float*)d_in[3],
                     (const float*)d_in[4], (const float*)d_in[5],
                     (const float*)d_in[6], (const float*)d_in[7],
                     (const float*)d_in[8], (const float*)d_in[9],
                     (float*)d_out);
}